// IsoNSProject_257698038373
// MI455X (gfx1250) — compile-verified
//
#include <hip/hip_runtime.h>
#include <math.h>

typedef float v2f __attribute__((ext_vector_type(2)));
typedef float v8f __attribute__((ext_vector_type(8)));

#define NN 2048
#define LDMAT 2048
#define MAT_ELEMS (2048u * 2048u)

// ---- GEMM tiling ----
#define BM 64
#define BN 64
#define BK 32
#define GEMM_THREADS 128

// LDS row strides (floats). All multiples of 4 floats -> 16B-aligned rows.
#define LDA0 36  // As[m][k] layout (TA==0): [BM][36]
#define LDA1 68  // As[k][m] layout (TA==1): [BK][68]
#define LDB0 68  // Bs[k][n] layout (TB==0): [BK][68]
#define LDB1 36  // Bs[n][k] layout (TB==1): [BN][36]
#define A_LDS_MAX 2304  // max(64*36, 32*68)
#define B_LDS_MAX 2304  // max(32*68, 64*36)

#if __has_builtin(__builtin_amdgcn_global_load_async_to_lds_b128)
#define HAVE_ASYNC_LDS 1
typedef int v4i __attribute__((vector_size(4 * sizeof(int))));
typedef __attribute__((address_space(1))) v4i g_v4i;   // global
typedef __attribute__((address_space(3))) v4i l_v4i;   // LDS
#else
#define HAVE_ASYNC_LDS 0
#endif

__device__ __forceinline__ void wait_async8() {
#if __has_builtin(__builtin_amdgcn_s_wait_asynccnt)
    __builtin_amdgcn_s_wait_asynccnt(8);
#else
    asm volatile("s_wait_asynccnt 0x8" ::: "memory");
#endif
}
__device__ __forceinline__ void wait_async0() {
#if __has_builtin(__builtin_amdgcn_s_wait_asynccnt)
    __builtin_amdgcn_s_wait_asynccnt(0);
#else
    asm volatile("s_wait_asynccnt 0x0" ::: "memory");
#endif
}

#if HAVE_ASYNC_LDS
__device__ __forceinline__ void async_cp16(const float* src, float* dst) {
    __builtin_amdgcn_global_load_async_to_lds_b128((g_v4i*)src, (l_v4i*)dst, 0, 0);
}
#endif

// C = op(A) * op(B) with optional epilogue.
// TA: 0 -> A as-is (MxK row-major), 1 -> A transposed (use A[k][m])
// TB: 0 -> B as-is (KxN row-major), 1 -> B transposed (use B[n][k])
// EPI: 0 -> C = P ; 1 -> C = 1.5*E - 0.5*P ; 2 -> C = P + addc
template<int TA, int TB, int EPI>
__global__ __launch_bounds__(GEMM_THREADS)
void gemm_wmma(const float* __restrict__ A, const float* __restrict__ B,
               const float* __restrict__ E, float* __restrict__ C, float addc) {
    __shared__ __align__(16) float As[2][A_LDS_MAX];
    __shared__ __align__(16) float Bs[2][B_LDS_MAX];

    const int tid  = threadIdx.x;
    const int lane = tid & 31;
    const int wave = tid >> 5;
    const int wm   = (wave >> 1) * 32;   // wave row offset within 64x64 tile
    const int wn   = (wave & 1) * 32;    // wave col offset
    const int mt   = blockIdx.y * BM;
    const int nt   = blockIdx.x * BN;

    // per-thread staging coordinates
    const int ar0 = tid >> 1, ac0 = (tid & 1) * 16;   // 64x32-tile pattern
    const int kr0 = tid >> 2, kc0 = (tid & 3) * 16;   // 32x64-tile pattern

    const v8f vzero = {0.f, 0.f, 0.f, 0.f, 0.f, 0.f, 0.f, 0.f};
    v8f acc[2][2] = {{vzero, vzero}, {vzero, vzero}};

    // ---- staging: issue loads for chunk at k0 into buffer buf ----
    auto stage = [&](int k0, int buf) {
        // A tile
        if (TA == 0) {
            const float* src = A + (size_t)(mt + ar0) * LDMAT + k0 + ac0;
            float* dst = &As[buf][ar0 * LDA0 + ac0];
            #pragma unroll
            for (int j = 0; j < 4; ++j) {
#if HAVE_ASYNC_LDS
                async_cp16(src + 4 * j, dst + 4 * j);
#else
                *(float4*)(dst + 4 * j) = *(const float4*)(src + 4 * j);
#endif
            }
        } else {
            const float* src = A + (size_t)(k0 + kr0) * LDMAT + mt + kc0;
            float* dst = &As[buf][kr0 * LDA1 + kc0];
            #pragma unroll
            for (int j = 0; j < 4; ++j) {
#if HAVE_ASYNC_LDS
                async_cp16(src + 4 * j, dst + 4 * j);
#else
                *(float4*)(dst + 4 * j) = *(const float4*)(src + 4 * j);
#endif
            }
        }
        // B tile
        if (TB == 0) {
            const float* src = B + (size_t)(k0 + kr0) * LDMAT + nt + kc0;
            float* dst = &Bs[buf][kr0 * LDB0 + kc0];
            #pragma unroll
            for (int j = 0; j < 4; ++j) {
#if HAVE_ASYNC_LDS
                async_cp16(src + 4 * j, dst + 4 * j);
#else
                *(float4*)(dst + 4 * j) = *(const float4*)(src + 4 * j);
#endif
            }
        } else {
            const float* src = B + (size_t)(nt + ar0) * LDMAT + k0 + ac0;
            float* dst = &Bs[buf][ar0 * LDB1 + ac0];
            #pragma unroll
            for (int j = 0; j < 4; ++j) {
#if HAVE_ASYNC_LDS
                async_cp16(src + 4 * j, dst + 4 * j);
#else
                *(float4*)(dst + 4 * j) = *(const float4*)(src + 4 * j);
#endif
            }
        }
    };

    // ---- compute one 32-deep K chunk from buffer buf ----
    auto compute = [&](int buf) {
        #pragma unroll
        for (int kk = 0; kk < BK; kk += 4) {
            // A frag (16x4 f32): lanes 0-15 hold K=kk..kk+1, lanes 16-31 hold K=kk+2..kk+3
            const int kb = kk + ((lane >> 4) << 1);
            const int ml = lane & 15;
            v2f afr[2], bfr[2];
            #pragma unroll
            for (int mi = 0; mi < 2; ++mi) {
                const int m = wm + mi * 16 + ml;
                if (TA == 0) {
                    afr[mi] = *(const v2f*)&As[buf][m * LDA0 + kb];
                } else {
                    afr[mi].x = As[buf][kb * LDA1 + m];
                    afr[mi].y = As[buf][(kb + 1) * LDA1 + m];
                }
            }
            // B frag (4x16 f32): VGPR0 = rows kk (lo lanes) / kk+2 (hi lanes), VGPR1 = +1
            #pragma unroll
            for (int ni = 0; ni < 2; ++ni) {
                const int n = wn + ni * 16 + ml;
                if (TB == 0) {
                    bfr[ni].x = Bs[buf][kb * LDB0 + n];
                    bfr[ni].y = Bs[buf][(kb + 1) * LDB0 + n];
                } else {
                    bfr[ni] = *(const v2f*)&Bs[buf][n * LDB1 + kb];
                }
            }
            #pragma unroll
            for (int mi = 0; mi < 2; ++mi)
                #pragma unroll
                for (int ni = 0; ni < 2; ++ni)
                    acc[mi][ni] = __builtin_amdgcn_wmma_f32_16x16x4_f32(
                        false, afr[mi], false, bfr[ni],
                        (short)0, acc[mi][ni], false, false);
        }
    };

#if HAVE_ASYNC_LDS
    // ---- double-buffered async pipeline ----
    stage(0, 0);
    int cur = 0;
    for (int k0 = 0; k0 < NN; k0 += BK) {
        if (k0 + BK < NN) {
            stage(k0 + BK, cur ^ 1);  // prefetch next chunk (8 async ops/wave)
            wait_async8();            // current chunk's 8 have landed
        } else {
            wait_async0();
        }
        __syncthreads();
        compute(cur);
        __syncthreads();
        cur ^= 1;
    }
#else
    // ---- synchronous single-buffer fallback ----
    for (int k0 = 0; k0 < NN; k0 += BK) {
        stage(k0, 0);
        __syncthreads();
        compute(0);
        __syncthreads();
    }
#endif

    // ---- epilogue: C layout VGPR r -> rows (r, r+8), cols = lane&15 ----
    const int rowoff = (lane >> 4) << 3;
    const int ncol   = lane & 15;
    #pragma unroll
    for (int mi = 0; mi < 2; ++mi) {
        #pragma unroll
        for (int ni = 0; ni < 2; ++ni) {
            #pragma unroll
            for (int r = 0; r < 8; ++r) {
                const int row = mt + wm + mi * 16 + r + rowoff;
                const int col = nt + wn + ni * 16 + ncol;
                const size_t idx = (size_t)row * LDMAT + col;
                float p = acc[mi][ni][r];
                if (EPI == 1)      p = 1.5f * E[idx] - 0.5f * p;
                else if (EPI == 2) p = p + addc;
                C[idx] = p;
            }
        }
    }
}

// ---- U repack: 2048x2047 -> zero-padded 2048x2048 ----
__global__ __launch_bounds__(256)
void repack_u(const float* __restrict__ U, float* __restrict__ Up) {
    const size_t idx = (size_t)blockIdx.x * 256 + threadIdx.x;
    const int i = (int)(idx >> 11);
    const int j = (int)(idx & 2047);
    Up[idx] = (j < 2047) ? U[(size_t)i * 2047 + j] : 0.f;
}

// ---- power-iteration helpers ----
__global__ __launch_bounds__(256)
void matvec_row(const float* __restrict__ A, const float* __restrict__ x,
                float* __restrict__ y) {
    __shared__ float red[256];
    const int row = blockIdx.x;
    const float* a = A + (size_t)row * LDMAT;
    float s = 0.f;
    for (int k = threadIdx.x; k < NN; k += 256) s += a[k] * x[k];
    red[threadIdx.x] = s;
    __syncthreads();
    for (int off = 128; off > 0; off >>= 1) {
        if (threadIdx.x < off) red[threadIdx.x] += red[threadIdx.x + off];
        __syncthreads();
    }
    if (threadIdx.x == 0) y[row] = red[0];
}

__global__ __launch_bounds__(256)
void matvec_col(const float* __restrict__ A, const float* __restrict__ x,
                float* __restrict__ y) {
    const int col = blockIdx.x * 256 + threadIdx.x;
    float s = 0.f;
    for (int i = 0; i < NN; ++i) s += A[(size_t)i * LDMAT + col] * x[i];
    y[col] = s;
}

__global__ __launch_bounds__(256)
void vec_norm(const float* __restrict__ x, float* __restrict__ nrm) {
    __shared__ float red[256];
    float s = 0.f;
    for (int i = threadIdx.x; i < NN; i += 256) { const float v = x[i]; s += v * v; }
    red[threadIdx.x] = s;
    __syncthreads();
    for (int off = 128; off > 0; off >>= 1) {
        if (threadIdx.x < off) red[threadIdx.x] += red[threadIdx.x + off];
        __syncthreads();
    }
    if (threadIdx.x == 0) nrm[0] = sqrtf(red[0]);
}

__global__ __launch_bounds__(256)
void vec_scale(const float* __restrict__ x, const float* __restrict__ nrm,
               float* __restrict__ y) {
    const int i = blockIdx.x * 256 + threadIdx.x;
    y[i] = x[i] / (nrm[0] + 1e-20f);
}

__global__ void finalize_scale(const float* __restrict__ nrm, float* __restrict__ scl) {
    // nrm = ||A^T A v|| with unit v  =>  sigma ~= sqrt(nrm)
    scl[0] = 1.0f / (sqrtf(nrm[0]) + 1e-8f);
}

__global__ __launch_bounds__(256)
void mat_scale(const float* __restrict__ A, const float* __restrict__ scl,
               float* __restrict__ X) {
    const size_t idx = (size_t)blockIdx.x * 256 + threadIdx.x;
    X[idx] = A[idx] * scl[0];
}

__global__ __launch_bounds__(256)
void ones_init(float* __restrict__ v) {
    v[blockIdx.x * 256 + threadIdx.x] = 1.0f;
}

extern "C" void kernel_launch(void* const* d_in, const int* in_sizes, int n_in,
                              void* d_out, int out_size, void* d_ws, size_t ws_size,
                              hipStream_t stream) {
    const float* H_raw = (const float*)d_in[0];   // 2048 x 2048
    const float* U_in  = (const float*)d_in[1];   // 2048 x 2047 (ld = 2047)
    float* out = (float*)d_out;                   // 2048 x 2048

    float* ws = (float*)d_ws;
    float* Up  = ws + 0 * (size_t)MAT_ELEMS;  // padded U, 2048x2048
    float* Amx = ws + 1 * (size_t)MAT_ELEMS;  // A = U^T H U (padded)
    float* Xa  = ws + 2 * (size_t)MAT_ELEMS;  // NS ping
    float* Xb  = ws + 3 * (size_t)MAT_ELEMS;  // NS pong
    float* Tm  = ws + 4 * (size_t)MAT_ELEMS;  // T1 / Y / T2 scratch
    float* v   = ws + 5 * (size_t)MAT_ELEMS;  // 2048
    float* w   = v + NN;                      // 2048
    float* u   = w + NN;                      // 2048
    float* nrm = u + NN;                      // scalar
    float* scl = nrm + 1;                     // scalar

    const dim3 ggrid(NN / BN, NN / BM);       // 32 x 32
    const dim3 gblock(GEMM_THREADS);
    const int  elem_blocks = (int)(MAT_ELEMS / 256);

    // 0) repack U with zero-padded extra column (keeps padding exactly 0 end-to-end)
    repack_u<<<elem_blocks, 256, 0, stream>>>(U_in, Up);

    // 1) A = U^T (H U)
    gemm_wmma<0, 0, 0><<<ggrid, gblock, 0, stream>>>(H_raw, Up, nullptr, Tm, 0.f);
    gemm_wmma<1, 0, 0><<<ggrid, gblock, 0, stream>>>(Up, Tm, nullptr, Amx, 0.f);

    // 2) spectral norm of A via power iteration on A^T A
    ones_init<<<NN / 256, 256, 0, stream>>>(v);
    for (int it = 0; it < 24; ++it) {
        matvec_row<<<NN, 256, 0, stream>>>(Amx, v, w);        // w = A v
        matvec_col<<<NN / 256, 256, 0, stream>>>(Amx, w, u);  // u = A^T w
        vec_norm<<<1, 256, 0, stream>>>(u, nrm);              // ||u||
        vec_scale<<<NN / 256, 256, 0, stream>>>(u, nrm, v);   // v = u / ||u||
    }
    finalize_scale<<<1, 1, 0, stream>>>(nrm, scl);

    // 3) X0 = A * (1/(sigma + 1e-8))
    mat_scale<<<elem_blocks, 256, 0, stream>>>(Amx, scl, Xa);

    // 4) 10 Newton-Schulz steps: Y = X^T X ; Xn = 1.5 X - 0.5 (X Y)
    float* Xcur = Xa;
    float* Xnext = Xb;
    for (int s = 0; s < 10; ++s) {
        gemm_wmma<1, 0, 0><<<ggrid, gblock, 0, stream>>>(Xcur, Xcur, nullptr, Tm, 0.f);
        gemm_wmma<0, 0, 1><<<ggrid, gblock, 0, stream>>>(Xcur, Tm, Xcur, Xnext, 0.f);
        float* t = Xcur; Xcur = Xnext; Xnext = t;
    }

    // 5) H = (1/n) * ones + U X U^T
    gemm_wmma<0, 0, 0><<<ggrid, gblock, 0, stream>>>(Up, Xcur, nullptr, Tm, 0.f);
    gemm_wmma<0, 1, 2><<<ggrid, gblock, 0, stream>>>(Tm, Up, nullptr, out, 1.0f / (float)NN);
}